// MambaBlock_67740224193299
// MI455X (gfx1250) — compile-verified
//
#include <hip/hip_runtime.h>
#include <hip/hip_bf16.h>
#include <math.h>

typedef __attribute__((ext_vector_type(2))) float v2f;
typedef __attribute__((ext_vector_type(8))) float v8f;

#define D_MODEL  4096
#define D_INNER  8192
#define D_STATE  64
#define PROJ_OUT 16513  // 2*D_INNER + 2*D_STATE + 1

__device__ __forceinline__ float siluf(float x) {
  return x / (1.0f + expf(-x));
}
__device__ __forceinline__ float softplusf(float x) {
  return fmaxf(x, 0.0f) + log1pf(expf(-fabsf(x)));
}

// ---------------------------------------------------------------------------
// Kernel 1: proj = W_in @ x   (16513 x 4096 GEMV, f32, via chained WMMA)
// One wave32 per 16-row tile. A = 16x4 tile of W (v2f per lane matches the ISA
// A layout exactly), B = x broadcast to all 16 columns. D column 0 -> lanes 0/16.
// ---------------------------------------------------------------------------
__global__ void __launch_bounds__(256) proj_gemv_kernel(
    const float* __restrict__ W, const float* __restrict__ x,
    float* __restrict__ proj) {
  __shared__ float xs[D_MODEL];
  for (int i = threadIdx.x * 4; i < D_MODEL; i += blockDim.x * 4)
    *(float4*)(xs + i) = *(const float4*)(x + i);
  __syncthreads();

  const int lane = threadIdx.x & 31;
  const int wave = threadIdx.x >> 5;
  const int tile = blockIdx.x * 8 + wave;
  const int full_tiles = PROJ_OUT >> 4;  // 1032

  if (tile < full_tiles) {
    const int m    = (tile << 4) + (lane & 15);
    const int koff = (lane >> 4) << 1;  // lanes 16..31 handle K+2,K+3
    const float* wrow = W + (size_t)m * D_MODEL + koff;
    const float* xp   = xs + koff;

    v8f acc0 = {}, acc1 = {}, acc2 = {}, acc3 = {};
    for (int k = 0; k < D_MODEL; k += 16) {
      __builtin_prefetch(wrow + k + 512, 0, 0);  // global_prefetch_b8, 2KB ahead
      v2f a0 = *(const v2f*)(wrow + k);
      v2f b0 = *(const v2f*)(xp + k);
      v2f a1 = *(const v2f*)(wrow + k + 4);
      v2f b1 = *(const v2f*)(xp + k + 4);
      v2f a2 = *(const v2f*)(wrow + k + 8);
      v2f b2 = *(const v2f*)(xp + k + 8);
      v2f a3 = *(const v2f*)(wrow + k + 12);
      v2f b3 = *(const v2f*)(xp + k + 12);
      acc0 = __builtin_amdgcn_wmma_f32_16x16x4_f32(false, a0, false, b0, (short)0, acc0, false, false);
      acc1 = __builtin_amdgcn_wmma_f32_16x16x4_f32(false, a1, false, b1, (short)0, acc1, false, false);
      acc2 = __builtin_amdgcn_wmma_f32_16x16x4_f32(false, a2, false, b2, (short)0, acc2, false, false);
      acc3 = __builtin_amdgcn_wmma_f32_16x16x4_f32(false, a3, false, b3, (short)0, acc3, false, false);
    }
    v8f acc = (acc0 + acc1) + (acc2 + acc3);
    const int mbase = tile << 4;
    if (lane == 0) {
#pragma unroll
      for (int r = 0; r < 8; ++r) proj[mbase + r] = acc[r];       // (M=r, N=0)
    } else if (lane == 16) {
#pragma unroll
      for (int r = 0; r < 8; ++r) proj[mbase + 8 + r] = acc[r];   // (M=8+r, N=0)
    }
  } else if (tile == full_tiles) {
    // trailing dt row (row 16512): plain wave32 dot product
    const int m = full_tiles << 4;
    const float* wrow = W + (size_t)m * D_MODEL;
    float sum = 0.0f;
    for (int k = lane; k < D_MODEL; k += 32) sum += wrow[k] * xs[k];
#pragma unroll
    for (int off = 16; off > 0; off >>= 1) sum += __shfl_xor(sum, off, 32);
    if (lane == 0) proj[m] = sum;
  }
}

// ---------------------------------------------------------------------------
// Kernel 2: SSM state update + gating. One wave32 per d-row of state (8192).
// Coalesced 256B float2 load of state[d,:], per-lane decay, shfl reduction.
// Writes h (output #2) and gated g (workspace) for the output GEMV.
// ---------------------------------------------------------------------------
__global__ void __launch_bounds__(256) ssm_kernel(
    const float* __restrict__ proj, const float* __restrict__ state,
    const float* __restrict__ A_log, const float* __restrict__ Dv,
    float* __restrict__ h_out, float* __restrict__ g) {
  const int lane = threadIdx.x & 31;
  const int wave = threadIdx.x >> 5;
  const int d    = blockIdx.x * 8 + wave;   // 0..8191
  const int s2   = lane << 1;               // each lane: 2 state columns

  const float dt  = softplusf(proj[2 * D_INNER + 2 * D_STATE]);  // proj[16512]
  const float zz  = proj[d];
  const float xsv = proj[D_INNER + d];
  const float sx  = siluf(xsv);

  const float2 al = *(const float2*)(A_log + s2);
  const float2 Bv = *(const float2*)(proj + 2 * D_INNER + s2);
  const float2 Cv = *(const float2*)(proj + 2 * D_INNER + D_STATE + s2);
  const float2 st = *(const float2*)(state + (size_t)d * D_STATE + s2);

  const float dec0 = expf(-expf(al.x) * dt);
  const float dec1 = expf(-expf(al.y) * dt);
  const float h0 = st.x * dec0 + sx * (dt * Bv.x);
  const float h1 = st.y * dec1 + sx * (dt * Bv.y);
  float2 hv; hv.x = h0; hv.y = h1;
  *(float2*)(h_out + (size_t)d * D_STATE + s2) = hv;

  float y = h0 * Cv.x + h1 * Cv.y;
#pragma unroll
  for (int off = 16; off > 0; off >>= 1) y += __shfl_xor(y, off, 32);
  if (lane == 0) {
    y += Dv[d] * sx;
    g[d] = y * siluf(zz);
  }
}

// ---------------------------------------------------------------------------
// Kernel 3: out_part = W_out @ g  (4096 x 8192 GEMV via WMMA, split-K = 4
// for occupancy; deterministic partial buffers, no float atomics).
// ---------------------------------------------------------------------------
__global__ void __launch_bounds__(256) out_gemv_kernel(
    const float* __restrict__ W, const float* __restrict__ g,
    float* __restrict__ part) {
  __shared__ float gs[D_INNER];
  for (int i = threadIdx.x * 4; i < D_INNER; i += blockDim.x * 4)
    *(float4*)(gs + i) = *(const float4*)(g + i);
  __syncthreads();

  const int lane  = threadIdx.x & 31;
  const int wave  = threadIdx.x >> 5;
  const int wid   = blockIdx.x * 8 + wave;  // 0..1023
  const int tile  = wid & 255;              // 256 row-tiles
  const int chunk = wid >> 8;               // 4 K-chunks
  const int klen  = D_INNER / 4;            // 2048
  const int kbase = chunk * klen;

  const int m    = (tile << 4) + (lane & 15);
  const int koff = (lane >> 4) << 1;
  const float* wrow = W + (size_t)m * D_INNER + kbase + koff;
  const float* gp   = gs + kbase + koff;

  v8f acc0 = {}, acc1 = {}, acc2 = {}, acc3 = {};
  for (int k = 0; k < klen; k += 16) {
    __builtin_prefetch(wrow + k + 512, 0, 0);
    v2f a0 = *(const v2f*)(wrow + k);
    v2f b0 = *(const v2f*)(gp + k);
    v2f a1 = *(const v2f*)(wrow + k + 4);
    v2f b1 = *(const v2f*)(gp + k + 4);
    v2f a2 = *(const v2f*)(wrow + k + 8);
    v2f b2 = *(const v2f*)(gp + k + 8);
    v2f a3 = *(const v2f*)(wrow + k + 12);
    v2f b3 = *(const v2f*)(gp + k + 12);
    acc0 = __builtin_amdgcn_wmma_f32_16x16x4_f32(false, a0, false, b0, (short)0, acc0, false, false);
    acc1 = __builtin_amdgcn_wmma_f32_16x16x4_f32(false, a1, false, b1, (short)0, acc1, false, false);
    acc2 = __builtin_amdgcn_wmma_f32_16x16x4_f32(false, a2, false, b2, (short)0, acc2, false, false);
    acc3 = __builtin_amdgcn_wmma_f32_16x16x4_f32(false, a3, false, b3, (short)0, acc3, false, false);
  }
  v8f acc = (acc0 + acc1) + (acc2 + acc3);
  float* dst = part + (size_t)chunk * D_MODEL + (tile << 4);
  if (lane == 0) {
#pragma unroll
    for (int r = 0; r < 8; ++r) dst[r] = acc[r];
  } else if (lane == 16) {
#pragma unroll
    for (int r = 0; r < 8; ++r) dst[8 + r] = acc[r];
  }
}

__global__ void __launch_bounds__(256) reduce_out_kernel(
    const float* __restrict__ part, float* __restrict__ out) {
  const int m = blockIdx.x * blockDim.x + threadIdx.x;
  if (m < D_MODEL)
    out[m] = (part[m] + part[D_MODEL + m]) +
             (part[2 * D_MODEL + m] + part[3 * D_MODEL + m]);
}

// ---------------------------------------------------------------------------
extern "C" void kernel_launch(void* const* d_in, const int* in_sizes, int n_in,
                              void* d_out, int out_size, void* d_ws, size_t ws_size,
                              hipStream_t stream) {
  const float* x     = (const float*)d_in[0];  // (1, 4096)
  const float* state = (const float*)d_in[1];  // (1, 8192, 64)
  const float* W_in  = (const float*)d_in[2];  // (16513, 4096)
  const float* A_log = (const float*)d_in[3];  // (64,)
  const float* Dv    = (const float*)d_in[4];  // (8192,)
  const float* W_out = (const float*)d_in[5];  // (4096, 8192)

  float* out = (float*)d_out;        // output 0: (4096,)
  float* h   = out + D_MODEL;        // output 1: (8192, 64)

  float* ws   = (float*)d_ws;
  float* proj = ws;                  // 16513 floats (padded region 16520)
  float* g    = ws + 16520;          // 8192 floats
  float* part = ws + 16520 + 8192;   // 4*4096 floats

  // 1) proj = W_in @ x : 1032 WMMA tiles + 1 tail row -> 1033 waves, 8/block
  proj_gemv_kernel<<<130, 256, 0, stream>>>(W_in, x, proj);
  // 2) state update + gate: one wave per d row (8192 waves)
  ssm_kernel<<<D_INNER / 8, 256, 0, stream>>>(proj, state, A_log, Dv, h, g);
  // 3) out = W_out @ g : 256 tiles x 4 K-chunks = 1024 waves
  out_gemv_kernel<<<128, 256, 0, stream>>>(W_out, g, part);
  // 4) deterministic split-K reduction
  reduce_out_kernel<<<(D_MODEL + 255) / 256, 256, 0, stream>>>(part, out);
}